// ResonantFFN_75892072121083
// MI455X (gfx1250) — compile-verified
//
#include <hip/hip_runtime.h>
#include <math.h>

typedef __attribute__((ext_vector_type(2))) float v2f;
typedef __attribute__((ext_vector_type(8))) float v8f;

#define BATCH  256
#define DMODEL 512
#define NNEUR  1024
#define LUTN   4096
#define BT     16
#define NT     16
#define WPAD   516   // 516 % 64 == 4 -> 16 rows map to 16 distinct LDS banks

// ---------------------------------------------------------------------------
// Phase 1: theta = x*(1/(1+|W|)) + B, LUT-interpolated sin/cos, reduce over d.
// One 256-thread block handles a 16(batch) x 16(neuron) tile; each thread owns
// one (b,n) pair and loops d=0..511 with per-thread accumulators.
// LUT lives in LDS as float2{sin,cos}[4097] (entry 4096 == entry 0 for wrap),
// so each element needs exactly two ds_load_b64 gathers.
// ---------------------------------------------------------------------------
__global__ __launch_bounds__(256) void resonant_phase1(
    const float* __restrict__ xr, const float* __restrict__ xi,
    const float* __restrict__ W,  const float* __restrict__ B,
    float* __restrict__ S /* (256 x 2048): [:,0:1024]=cos_sum, [:,1024:]=sin_sum */)
{
  __shared__ float2 lut[LUTN + 1];       // 32776 B
  __shared__ float  xs[BT * DMODEL];     // 32768 B
  __shared__ float  ws[NT * WPAD];       // 33024 B  (invw = 1/(1+|W|))
  __shared__ float  bs[NT * WPAD];       // 33024 B

  const int tid   = threadIdx.x;
  const int ntile = blockIdx.x;  // 0..63
  const int btile = blockIdx.y;  // 0..15

  // Build the interpolation table (grid = i * 2pi/4096, f32 like the reference).
  for (int i = tid; i < LUTN + 1; i += 256) {
    int   j   = i & (LUTN - 1);
    float ang = (float)j * 0.0015339807878856412f;   // 2pi/4096
    lut[i] = make_float2(sinf(ang), cosf(ang));
  }
  // Stage x tile (x = x_real + x_imag), coalesced.
  for (int i = tid; i < BT * DMODEL; i += 256) {
    int r = i >> 9, c = i & (DMODEL - 1);
    int g = (btile * BT + r) * DMODEL + c;
    xs[i] = xr[g] + xi[g];
  }
  // Stage reciprocal-wavelength and bias tiles with bank-conflict padding.
  for (int i = tid; i < NT * DMODEL; i += 256) {
    int r = i >> 9, c = i & (DMODEL - 1);
    int g = (ntile * NT + r) * DMODEL + c;
    ws[r * WPAD + c] = 1.0f / (1.0f + fabsf(W[g]));
    bs[r * WPAD + c] = B[g];
  }
  __syncthreads();

  const int ni = tid & 15;        // neuron within tile
  const int bi = tid >> 4;        // batch row within tile
  const float* xrow = &xs[bi * DMODEL];   // broadcast across 16 lanes
  const float* wrow = &ws[ni * WPAD];
  const float* brow = &bs[ni * WPAD];

  float accs = 0.0f, accc = 0.0f;
#pragma unroll 4
  for (int d = 0; d < DMODEL; ++d) {
    float theta = fmaf(xrow[d], wrow[d], brow[d]);
    float pos   = theta * 651.8986469044033f;        // * 4096/2pi
    float fl    = floorf(pos);
    float frac  = pos - fl;                          // in [0,1)
    int   i0    = ((int)fl) & (LUTN - 1);            // == python mod for pow2
    float2 e0 = lut[i0];
    float2 e1 = lut[i0 + 1];                         // wrap entry handles 4095->0
    accs += fmaf(frac, e1.x - e0.x, e0.x);           // sin interp
    accc += fmaf(frac, e1.y - e0.y, e0.y);           // cos interp
  }

  const int bg = btile * BT + bi;
  const int ng = ntile * NT + ni;
  S[bg * 2048 + ng]        = accc;   // cos_sum
  S[bg * 2048 + 1024 + ng] = accs;   // sin_sum
}

// ---------------------------------------------------------------------------
// Phase 2: out = cos_sum @ Pc^T + sin_sum @ Ps^T, fused SiLU epilogue.
// Full-precision f32 WMMA (V_WMMA_F32_16X16X4_F32). One wave per 16x16 output
// tile; 8 waves per block. A-frag: lane<16 holds M=lane K={0,1}; lanes>=16
// hold M=lane-16 K={2,3} (ISA 16x4 f32 layout); B-frag mirrored for 4x16.
// ---------------------------------------------------------------------------
__global__ __launch_bounds__(256) void resonant_phase2(
    const float* __restrict__ S,  const float* __restrict__ Pc,
    const float* __restrict__ Ps, float* __restrict__ out)
{
  const int lane = threadIdx.x & 31;
  const int wave = threadIdx.x >> 5;
  const int tile = blockIdx.x * 8 + wave;  // 512 tiles total
  const int tm   = tile >> 5;              // batch-tile   0..15
  const int tn   = tile & 31;              // dmodel-tile  0..31
  const int idx  = lane & 15;
  const int hv   = lane >> 4;              // half-wave -> K offset (0 or 2)

  const int arow = tm * 16 + idx;          // batch row for A fragment
  const int bcol = tn * 16 + idx;          // output column = row of Pc/Ps

  const float* aptrC = &S[arow * 2048 + hv * 2];          // cos panel
  const float* aptrS = aptrC + 1024;                      // sin panel
  const float* bptrC = &Pc[bcol * 1024 + hv * 2];
  const float* bptrS = &Ps[bcol * 1024 + hv * 2];

  v8f acc = {0.f, 0.f, 0.f, 0.f, 0.f, 0.f, 0.f, 0.f};

  for (int k = 0; k < 1024; k += 4) {
    v2f a = *(const v2f*)(aptrC + k);
    v2f b = *(const v2f*)(bptrC + k);
    acc = __builtin_amdgcn_wmma_f32_16x16x4_f32(false, a, false, b,
                                                (short)0, acc, false, false);
  }
  for (int k = 0; k < 1024; k += 4) {
    v2f a = *(const v2f*)(aptrS + k);
    v2f b = *(const v2f*)(bptrS + k);
    acc = __builtin_amdgcn_wmma_f32_16x16x4_f32(false, a, false, b,
                                                (short)0, acc, false, false);
  }

  // C/D layout: VGPR g holds M = g + 8*(lane>=16), N = lane&15. Fused SiLU.
#pragma unroll
  for (int g = 0; g < 8; ++g) {
    int   m = tm * 16 + g + hv * 8;
    int   n = tn * 16 + idx;
    float v = acc[g];
    out[m * DMODEL + n] = v / (1.0f + expf(-v));   // silu
  }
}

extern "C" void kernel_launch(void* const* d_in, const int* in_sizes, int n_in,
                              void* d_out, int out_size, void* d_ws, size_t ws_size,
                              hipStream_t stream) {
  const float* xr = (const float*)d_in[0];
  const float* xi = (const float*)d_in[1];
  const float* W  = (const float*)d_in[2];
  const float* B  = (const float*)d_in[3];
  const float* Pc = (const float*)d_in[4];
  const float* Ps = (const float*)d_in[5];
  float* S = (float*)d_ws;   // 256*2048 f32 = 2 MB scratch panel [cos|sin]

  hipLaunchKernelGGL(resonant_phase1, dim3(64, 16), dim3(256), 0, stream,
                     xr, xi, W, B, S);
  hipLaunchKernelGGL(resonant_phase2, dim3(64), dim3(256), 0, stream,
                     S, Pc, Ps, (float*)d_out);
}